// GINNet_15092515078531
// MI455X (gfx1250) — compile-verified
//
#include <hip/hip_runtime.h>

typedef __attribute__((ext_vector_type(2))) float v2f;
typedef __attribute__((ext_vector_type(8))) float v8f;

#define DIM 128

// ---------------------------------------------------------------------------
// Zero a buffer (float4 granularity; all sizes here are multiples of 4).
// ---------------------------------------------------------------------------
__global__ void zero4_kernel(float4* __restrict__ p, int n4) {
    int i = blockIdx.x * blockDim.x + threadIdx.x;
    int stride = gridDim.x * blockDim.x;
    for (; i < n4; i += stride) p[i] = make_float4(0.f, 0.f, 0.f, 0.f);
}

// ---------------------------------------------------------------------------
// Edge scatter: agg[dst] += x[src].  One wave32 per edge; each lane moves a
// float4 (32 lanes * 4 = 128 features) and issues 4 native f32 atomics
// (global_atomic_add_f32 via unsafeAtomicAdd). Both x and agg are L2-resident
// (51.2 MB each << 192 MB L2), so this runs at L2/atomic rate.
// ---------------------------------------------------------------------------
__global__ void scatter_kernel(const float* __restrict__ X,
                               const int* __restrict__ ei,
                               float* __restrict__ agg,
                               int E) {
    const int wave = threadIdx.x >> 5;
    const int lane = threadIdx.x & 31;
    const int e = blockIdx.x * (blockDim.x >> 5) + wave;
    if (e >= E) return;
    const int src = ei[e];         // edge_index[0][e]
    const int dst = ei[E + e];     // edge_index[1][e]
    const float4 v = ((const float4*)(X + (long long)src * DIM))[lane];
    float* a = agg + (long long)dst * DIM + lane * 4;
    unsafeAtomicAdd(a + 0, v.x);
    unsafeAtomicAdd(a + 1, v.y);
    unsafeAtomicAdd(a + 2, v.z);
    unsafeAtomicAdd(a + 3, v.w);
}

// ---------------------------------------------------------------------------
// Fused GIN GEMM:  Y = act( (X [+ AGG]) @ W + bias ),  W is [128,128] row-major.
// Block = 256 threads = 8 waves; block computes a 16x128 output strip
// (N = 100000 is a multiple of 16, so no edge guard needed).
//   - A-tile (16x128 f32, 8 KB) staged in LDS with the GIN combine fused in.
//   - wave w computes columns [16w, 16w+16) with 32 x V_WMMA_F32_16X16X4_F32.
// A-fragment layout (16x4 f32): lanes 0-15 -> rows, VGPR0/1 = K0/K1;
// lanes 16-31 -> same rows, K2/K3.  B symmetric with columns in lanes.
// C/D layout: VGPR v, lane l : row = v + 8*(l>=16), col = l&15.
// ---------------------------------------------------------------------------
template <bool RELU, bool HASAGG>
__global__ void gin_gemm_kernel(const float* __restrict__ X,
                                const float* __restrict__ AGG,
                                const float* __restrict__ W,
                                const float* __restrict__ bias,
                                float* __restrict__ Y) {
    __shared__ float Atile[16 * DIM];

    const int r0 = blockIdx.x * 16;

    // Stage A tile (fuse z = x + agg; GIN eps = 0).
    for (int idx = threadIdx.x; idx < 16 * DIM; idx += blockDim.x) {
        const int row = idx >> 7;      // / 128
        const int col = idx & (DIM - 1);
        const long long g = (long long)(r0 + row) * DIM + col;
        float v = X[g];
        if (HASAGG) v += AGG[g];
        Atile[idx] = v;
    }
    __syncthreads();

    const int wave = threadIdx.x >> 5;
    const int lane = threadIdx.x & 31;
    const int half = lane >> 4;        // 0: K{0,1}, 1: K{2,3}
    const int r    = lane & 15;        // A row within strip / B column within slab
    const int col  = wave * 16 + r;    // global output column

    v8f acc = {};
#pragma unroll
    for (int k0 = 0; k0 < DIM; k0 += 4) {
        const int ks = k0 + half * 2;
        v2f a, b;
        a.x = Atile[r * DIM + ks];     // contiguous pair -> ds_load_b64
        a.y = Atile[r * DIM + ks + 1];
        b.x = W[ks * DIM + col];       // row-major weights, lanes 0-15 coalesce
        b.y = W[(ks + 1) * DIM + col];
        acc = __builtin_amdgcn_wmma_f32_16x16x4_f32(
            /*neg_a=*/false, a, /*neg_b=*/false, b,
            /*c_mod=*/(short)0, acc, /*reuse_a=*/false, /*reuse_b=*/false);
    }

    const float bc = bias[col];
#pragma unroll
    for (int v = 0; v < 8; ++v) {
        float val = acc[v] + bc;
        if (RELU) val = fmaxf(val, 0.f);
        const int row = r0 + v + half * 8;
        Y[(long long)row * DIM + col] = val;
    }
}

// ---------------------------------------------------------------------------
// Head: out[n, 0:2] = x3[n, :] @ Wf2 (128x2) + bf2.  Wf2 cached in LDS.
// ---------------------------------------------------------------------------
__global__ void head_kernel(const float* __restrict__ X,
                            const float* __restrict__ Wf2,
                            const float* __restrict__ bf2,
                            float* __restrict__ out, int n) {
    __shared__ float w[DIM * 2];
    for (int i = threadIdx.x; i < DIM * 2; i += blockDim.x) w[i] = Wf2[i];
    __syncthreads();

    const int node = blockIdx.x * blockDim.x + threadIdx.x;
    if (node >= n) return;

    const float4* xr = (const float4*)(X + (long long)node * DIM);
    float a0 = bf2[0], a1 = bf2[1];
#pragma unroll
    for (int k4 = 0; k4 < DIM / 4; ++k4) {
        const float4 x = xr[k4];
        const int k = k4 * 4;
        a0 += x.x * w[(k + 0) * 2];  a1 += x.x * w[(k + 0) * 2 + 1];
        a0 += x.y * w[(k + 1) * 2];  a1 += x.y * w[(k + 1) * 2 + 1];
        a0 += x.z * w[(k + 2) * 2];  a1 += x.z * w[(k + 2) * 2 + 1];
        a0 += x.w * w[(k + 3) * 2];  a1 += x.w * w[(k + 3) * 2 + 1];
    }
    out[node * 2 + 0] = a0;
    out[node * 2 + 1] = a1;
}

// ---------------------------------------------------------------------------
// Launch: zero -> scatter -> gemm(agg) -> gemm   (layer 1)
//         zero -> scatter -> gemm(agg) -> gemm   (layer 2)
//         gemm (Wf1) -> head (Wf2)
// ---------------------------------------------------------------------------
extern "C" void kernel_launch(void* const* d_in, const int* in_sizes, int n_in,
                              void* d_out, int out_size, void* d_ws, size_t ws_size,
                              hipStream_t stream) {
    const float* h   = (const float*)d_in[0];
    // d_in[1] = bf (unused by reference)
    const int*   ei  = (const int*)d_in[2];
    // d_in[3] = edge_weight (unused by reference)
    const float* W1a = (const float*)d_in[4];
    const float* b1a = (const float*)d_in[5];
    const float* W1b = (const float*)d_in[6];
    const float* b1b = (const float*)d_in[7];
    const float* W2a = (const float*)d_in[8];
    const float* b2a = (const float*)d_in[9];
    const float* W2b = (const float*)d_in[10];
    const float* b2b = (const float*)d_in[11];
    const float* Wf1 = (const float*)d_in[12];
    const float* bf1 = (const float*)d_in[13];
    const float* Wf2 = (const float*)d_in[14];
    const float* bf2 = (const float*)d_in[15];

    const int N = in_sizes[0] / DIM;     // 100000
    const int E = in_sizes[2] / 2;       // 3200000

    const size_t fsz = (size_t)N * DIM;  // floats per node matrix
    float* bufA = (float*)d_ws;          // agg scratch
    float* bufB = bufA + fsz;            // ping
    float* bufC = bufB + fsz;            // pong

    const dim3 blk(256);
    const int gemm_grid = (N + 15) / 16;
    const int scat_grid = (E + 7) / 8;               // one wave32 per edge
    const int zero_grid = (int)((fsz / 4 + 255) / 256);
    const int head_grid = (N + 255) / 256;

    // ---- Layer 1: x1 = ReLU( MLP1( h + segsum(h[src] -> dst) ) ) ----
    zero4_kernel<<<zero_grid, blk, 0, stream>>>((float4*)bufA, (int)(fsz / 4));
    scatter_kernel<<<scat_grid, blk, 0, stream>>>(h, ei, bufA, E);
    gin_gemm_kernel<true, true ><<<gemm_grid, blk, 0, stream>>>(h,    bufA, W1a, b1a, bufB);
    gin_gemm_kernel<true, false><<<gemm_grid, blk, 0, stream>>>(bufB, nullptr, W1b, b1b, bufC);

    // ---- Layer 2: x2 = ReLU( MLP2( x1 + segsum(x1[src] -> dst) ) ) ----
    zero4_kernel<<<zero_grid, blk, 0, stream>>>((float4*)bufA, (int)(fsz / 4));
    scatter_kernel<<<scat_grid, blk, 0, stream>>>(bufC, ei, bufA, E);
    gin_gemm_kernel<true, true ><<<gemm_grid, blk, 0, stream>>>(bufC, bufA, W2a, b2a, bufB);
    gin_gemm_kernel<true, false><<<gemm_grid, blk, 0, stream>>>(bufB, nullptr, W2b, b2b, bufC);

    // ---- Heads ----
    gin_gemm_kernel<true, false><<<gemm_grid, blk, 0, stream>>>(bufC, nullptr, Wf1, bf1, bufB);
    head_kernel<<<head_grid, blk, 0, stream>>>(bufB, Wf2, bf2, (float*)d_out, N);
}